// spn_block_16673063043386
// MI455X (gfx1250) — compile-verified
//
#include <hip/hip_runtime.h>
#include <stdint.h>

// ---------------------------------------------------------------------------
// SPN left->right scan on gfx1250 (MI455X).
// Memory-bound (320 MB total, ~13.7 us floor at 23.3 TB/s). Strategy:
//  * one workgroup per (b,c) plane, 256 threads (8 wave32s), thread i = row i
//  * TDM (tensor_load_to_lds) stages 256x16 column tiles of x,G1,G2,G3 into
//    LDS with a 1-DWORD pad per 16-DWORD row (17-float row stride ->
//    bank-conflict-free column reads), double buffered, issued by wave 0 and
//    synchronized with s_wait_tensorcnt thresholds to overlap DMA + compute.
//  * scan state h kept in a ping-pong LDS array with zero halos; one
//    workgroup barrier per scan step (W=256 steps).
//  * output tiles staged in LDS and written back with tensor_store_from_lds.
// ---------------------------------------------------------------------------

typedef unsigned int u32x4 __attribute__((ext_vector_type(4)));
typedef int          i32x4 __attribute__((ext_vector_type(4)));
typedef int          i32x8 __attribute__((ext_vector_type(8)));

static constexpr int Hh   = 256;      // rows (H)
static constexpr int Ww   = 256;      // cols (W)
static constexpr int TW   = 16;       // tile width along W
static constexpr int NT   = Ww / TW;  // 16 tiles
static constexpr int ROWP = TW + 1;   // padded LDS row stride (floats)

// Issue a 2D TDM transfer. Descriptor layout per CDNA5 ISA.
__device__ __forceinline__ void tdm_issue_2d(uint32_t lds_addr, uint64_t gaddr,
                                             uint32_t tdim0, uint32_t tdim1,
                                             uint64_t stride0,
                                             uint32_t tile0, uint32_t tile1,
                                             bool pad, bool is_store)
{
  // Group 0: count=1 | lds_addr | global_addr[56:0] | type=2 ("image")
  u32x4 g0 = {
    1u,
    lds_addr,
    (uint32_t)gaddr,
    (uint32_t)((gaddr >> 32) & 0x01FFFFFFu) | (2u << 30)
  };
  // Group 1: data_size=4B; optional pad (interval=16 DW -> enc 3, amount=1 DW
  // -> enc 0); tensor_dim0/1; tile_dim0/1; tensor_dim0_stride. 2D only:
  // tile_dim2 = 0, tensor_dim1_stride = 0.
  uint32_t w0 = (2u << 16);
  if (pad) w0 |= (1u << 20) | (3u << 22);
  i32x8 g1 = {
    (int)w0,
    (int)((tdim0 & 0xFFFFu) << 16),
    (int)(((tdim0 >> 16) & 0xFFFFu) | ((tdim1 & 0xFFFFu) << 16)),
    (int)(((tdim1 >> 16) & 0xFFFFu) | ((tile0 & 0xFFFFu) << 16)),
    (int)(tile1 & 0xFFFFu),
    (int)(uint32_t)(stride0 & 0xFFFFFFFFu),
    (int)((uint32_t)(stride0 >> 32) & 0xFFFFu),
    0
  };
  i32x4 g2 = {0, 0, 0, 0};   // 2D: dims 2..4 unused
  i32x4 g3 = {0, 0, 0, 0};
#if defined(__clang_major__) && (__clang_major__ >= 23)
  i32x8 gz = {0, 0, 0, 0, 0, 0, 0, 0};
  if (is_store) __builtin_amdgcn_tensor_store_from_lds(g0, g1, g2, g3, gz, 0);
  else          __builtin_amdgcn_tensor_load_to_lds (g0, g1, g2, g3, gz, 0);
#else
  if (is_store) __builtin_amdgcn_tensor_store_from_lds(g0, g1, g2, g3, 0);
  else          __builtin_amdgcn_tensor_load_to_lds (g0, g1, g2, g3, 0);
#endif
}

__global__ __launch_bounds__(256, 1)
void spn_scan_kernel(const float* __restrict__ x,
                     const float* __restrict__ G1,
                     const float* __restrict__ G2,
                     const float* __restrict__ G3,
                     float* __restrict__ out)
{
  // 2 (double buffer) x 4 tensors x 256 rows x 17 padded floats = 136 KB
  __shared__ float s_in[2][4][Hh * ROWP];
  // output staging, unpadded (TDM store cannot de-pad): 32 KB
  __shared__ float s_out[2][Hh * TW];
  // ping-pong scan state with zero halos at [0] and [H+1]
  __shared__ float s_h[2][Hh + 2];

  const int tid   = threadIdx.x;
  const int wid   = tid >> 5;                       // wave id (wave32)
  const int plane = blockIdx.x;                     // (b*C + c)
  const uint64_t plane_off = (uint64_t)plane * (uint64_t)(Hh * Ww);

  // zero both h buffers (initial state + permanent halos)
  for (int j = tid; j < 2 * (Hh + 2); j += 256) ((float*)s_h)[j] = 0.0f;

  const float* bases[4] = {x, G1, G2, G3};

  // Prologue: wave 0 issues TDM loads for tile 0 into buffer 0.
  if (wid == 0) {
#pragma unroll
    for (int t = 0; t < 4; ++t) {
      uint64_t ga = (uint64_t)(uintptr_t)(bases[t] + plane_off);
      uint32_t la = (uint32_t)(uintptr_t)&s_in[0][t][0];
      tdm_issue_2d(la, ga, Ww, Hh, (uint64_t)Ww, TW, Hh,
                   /*pad=*/true, /*store=*/false);
    }
  }

  int q = 0;                                        // h ping-pong selector
  for (int k = 0; k < NT; ++k) {
    const int p = k & 1;

    if (wid == 0) {
      // prefetch tile k+1 into the other input buffer
      if (k + 1 < NT) {
#pragma unroll
        for (int t = 0; t < 4; ++t) {
          uint64_t ga = (uint64_t)(uintptr_t)
              (bases[t] + plane_off + (uint64_t)((k + 1) * TW));
          uint32_t la = (uint32_t)(uintptr_t)&s_in[p ^ 1][t][0];
          tdm_issue_2d(la, ga, Ww, Hh, (uint64_t)Ww, TW, Hh, true, false);
        }
      }
      // TDM ops complete in issue order per wave. Leaving the newest
      // {loads(k+1)[4], store(k-1)[1]} outstanding guarantees loads(k) and
      // store(k-2) (whose out buffer we are about to rewrite) are done.
      if (k == 0)          __builtin_amdgcn_s_wait_tensorcnt(4);
      else if (k + 1 < NT) __builtin_amdgcn_s_wait_tensorcnt(5);
      else                 __builtin_amdgcn_s_wait_tensorcnt(1);
    }
    __syncthreads();   // tile k visible to all waves; all prior reads retired

    const float* __restrict__ xin  = &s_in[p][0][0];
    const float* __restrict__ g1in = &s_in[p][1][0];
    const float* __restrict__ g2in = &s_in[p][2][0];
    const float* __restrict__ g3in = &s_in[p][3][0];
    const int rb = tid * ROWP;     // 17-float stride -> conflict-free columns

#pragma unroll 4
    for (int lt = 0; lt < TW; ++lt) {
      float xv = xin [rb + lt];
      float g1 = g1in[rb + lt];
      float g2 = g2in[rb + lt];
      float g3 = g3in[rb + lt];
      float up  = s_h[q][tid];         // h_prev[i-1] (halo -> 0)
      float mid = s_h[q][tid + 1];     // h_prev[i]
      float dn  = s_h[q][tid + 2];     // h_prev[i+1] (halo -> 0)

      // gate normalization: divide by |g1|+|g2|+|g3| only when sum >= 1
      float sa = __builtin_fabsf(g1) + __builtin_fabsf(g2) + __builtin_fabsf(g3);
      float scale = (sa >= 1.0f) ? __builtin_amdgcn_rcpf(sa) : 1.0f;
      g1 *= scale; g2 *= scale; g3 *= scale;

      float h = (1.0f - g1 - g2 - g3) * xv + g1 * up + g2 * mid + g3 * dn;

      s_h[q ^ 1][tid + 1]     = h;     // next-step state
      s_out[p][tid * TW + lt] = h;     // output staging
      __syncthreads();                 // one barrier per scan step
      q ^= 1;
    }

    // write tile k back via TDM (all waves' LDS writes retired by barrier)
    if (wid == 0) {
      uint64_t ga = (uint64_t)(uintptr_t)(out + plane_off + (uint64_t)(k * TW));
      uint32_t la = (uint32_t)(uintptr_t)&s_out[p][0];
      tdm_issue_2d(la, ga, Ww, Hh, (uint64_t)Ww, TW, Hh,
                   /*pad=*/false, /*store=*/true);
    }
  }

  if (wid == 0) __builtin_amdgcn_s_wait_tensorcnt(0);  // drain final store
}

extern "C" void kernel_launch(void* const* d_in, const int* in_sizes, int n_in,
                              void* d_out, int out_size, void* d_ws, size_t ws_size,
                              hipStream_t stream) {
  (void)n_in; (void)out_size; (void)d_ws; (void)ws_size;
  const float* x  = (const float*)d_in[0];
  const float* G1 = (const float*)d_in[1];
  const float* G2 = (const float*)d_in[2];
  const float* G3 = (const float*)d_in[3];
  float* out = (float*)d_out;

  const int planes = in_sizes[0] / (Hh * Ww);   // B*C = 256
  hipLaunchKernelGGL(spn_scan_kernel, dim3(planes), dim3(256), 0, stream,
                     x, G1, G2, G3, out);
}